// GraphObservationEncoder_21492016349945
// MI455X (gfx1250) — compile-verified
//
#include <hip/hip_runtime.h>
#include <math.h>
#include <stdint.h>

// ---------------------------------------------------------------------------
// GAT encoder forward for MI455X (gfx1250, wave32).
//  - GEMMs: V_WMMA_F32_16X16X4_F32, one wave = 16 rows x full Nout (8 tiles),
//    W staged into LDS via GLOBAL_LOAD_ASYNC_TO_LDS_B128 (ASYNCcnt path).
//    B fragments all fetched into registers first so LDS loads batch ahead of
//    the WMMA burst instead of serializing on s_wait_dscnt.
//  - Edge phase: wave32-native gather/scatter with f32 atomics (L2-resident).
// ---------------------------------------------------------------------------

#define GN      50000
#define GE      1000000
#define F_IN    32
#define HID     128
#define OUTD    64
#define HEADS   4
#define HD      32
#define NLAYERS 3

typedef float v2f __attribute__((ext_vector_type(2)));
typedef float v8f __attribute__((ext_vector_type(8)));

__device__ __forceinline__ float gelu_exact(float x) {
    return 0.5f * x * (1.0f + erff(x * 0.70710678118654752f));
}

// ---------------------------------------------------------------------------
// C[M x NT*16] = act(A[M x K] @ W[NT*16 x K]^T + b)
// Block = 4 waves = 64 rows. W (NT*16 x K) staged into LDS once per block via
// async-to-LDS copies (padded row stride K+4 floats -> conflict-free b64
// B-fragment reads). Each wave computes 16 rows x NT*16 cols: one A fragment
// per k-step feeds NT WMMAs; all NT B fragments are fetched before the WMMA
// burst so the LDS loads pipeline.
// Requires M % 16 == 0, K % 4 == 0.
// ---------------------------------------------------------------------------
template <int NT>
__global__ void wmma_gemm_lds(const float* __restrict__ A,
                              const float* __restrict__ W,
                              const float* __restrict__ bias,
                              float* __restrict__ C,
                              int M, int K, int act_gelu)
{
    constexpr int NOUT = NT * 16;
    const int KP = K + 4;                       // padded LDS row stride
    extern __shared__ float smem[];             // NOUT * KP floats

    // ---- stage W into LDS with CDNA5 async copies (16 B per lane per op) --
    const int tid = threadIdx.y * 32 + threadIdx.x;
    const int chunksPerRow = K >> 2;            // 4-float chunks
    const int totalChunks  = NOUT * chunksPerRow;
    const uint32_t lds0 = (uint32_t)(uintptr_t)smem;   // LDS aperture: low 32b
    for (int c = tid; c < totalChunks; c += 128) {
        const int row = c / chunksPerRow;
        const int c4  = (c - row * chunksPerRow) << 2;
        const uint32_t ldsoff = lds0 + (uint32_t)(row * KP + c4) * 4u;
        const uint64_t gaddr  = (uint64_t)(uintptr_t)(W + (size_t)row * K + c4);
        asm volatile("global_load_async_to_lds_b128 %0, %1, off"
                     :: "v"(ldsoff), "v"(gaddr) : "memory");
    }
    asm volatile("s_wait_asynccnt 0x0" ::: "memory");   // own copies done
    __syncthreads();                                    // all waves' copies

    // ---- compute -----------------------------------------------------------
    const int lane = threadIdx.x;
    const int mr = lane & 15;                  // M (A) / N (B,C) in tile
    const int hi = lane >> 4;                  // lane half -> K sub-pair
    const int row0 = blockIdx.x * 64 + threadIdx.y * 16;
    if (row0 >= M) return;                     // wave-uniform (tail block)

    v8f acc[NT];
    const v8f vzero = {0.f, 0.f, 0.f, 0.f, 0.f, 0.f, 0.f, 0.f};
#pragma unroll
    for (int t = 0; t < NT; ++t) acc[t] = vzero;

    // LDS base for this lane's B fragments (8-byte aligned: KP even, 2*hi even)
    const float* __restrict__ wbase = smem + mr * KP + 2 * hi;
    const float* __restrict__ ap    = A + (size_t)(row0 + mr) * K + 2 * hi;

    for (int k0 = 0; k0 < K; k0 += 4) {
        v2f a;
        a.x = ap[k0];                          // A[row0+mr][k0+2*hi]
        a.y = ap[k0 + 1];

        v2f bfrag[NT];                         // fetch all B fragments first
#pragma unroll
        for (int t = 0; t < NT; ++t)
            bfrag[t] = *(const v2f*)(wbase + t * 16 * KP + k0);

#pragma unroll
        for (int t = 0; t < NT; ++t)
            acc[t] = __builtin_amdgcn_wmma_f32_16x16x4_f32(
                         false, a, false, bfrag[t], (short)0, acc[t],
                         false, false);
    }

#pragma unroll
    for (int t = 0; t < NT; ++t) {
        const int n = t * 16 + mr;
        const float bn = bias[n];
#pragma unroll
        for (int r = 0; r < 8; ++r) {
            float v = acc[t][r] + bn;
            if (act_gelu) v = gelu_exact(v);
            C[(size_t)(row0 + r + 8 * hi) * NOUT + n] = v;
        }
    }
}

// ---------------------------------------------------------------------------
// Per-(node, head) attention logits: a = sum_d x[n][h*32+d] * attn[h][d]
// ---------------------------------------------------------------------------
__global__ void attn_logits_kernel(const float* __restrict__ xsrc,
                                   const float* __restrict__ xdst,
                                   const float* __restrict__ attn_src,
                                   const float* __restrict__ attn_dst,
                                   float* __restrict__ a_src,
                                   float* __restrict__ a_dst, int n)
{
    int idx = blockIdx.x * blockDim.x + threadIdx.x;
    if (idx >= n * HEADS) return;
    const int h    = idx & (HEADS - 1);
    const int node = idx >> 2;
    const float* xs = xsrc + (size_t)node * HID + h * HD;
    const float* xd = xdst + (size_t)node * HID + h * HD;
    const float* as = attn_src + h * HD;
    const float* ad = attn_dst + h * HD;
    float ss = 0.f, sd = 0.f;
#pragma unroll
    for (int i = 0; i < HD; ++i) { ss += xs[i] * as[i]; sd += xd[i] * ad[i]; }
    a_src[idx] = ss;
    a_dst[idx] = sd;
}

__global__ void zero_f32(float* __restrict__ p, size_t n)
{
    size_t i = (size_t)blockIdx.x * blockDim.x + threadIdx.x;
    if (i < n) p[i] = 0.f;
}

// ---------------------------------------------------------------------------
// Per-edge attention numerator + atomic denominator segment-sum.
// ---------------------------------------------------------------------------
__global__ void edge_alpha_kernel(const int* __restrict__ src,
                                  const int* __restrict__ dst,
                                  const float* __restrict__ edge_attr,
                                  const float* __restrict__ eW,   // (HEADS,1)
                                  const float* __restrict__ eb,   // (HEADS)
                                  const float* __restrict__ a_src,
                                  const float* __restrict__ a_dst,
                                  float* __restrict__ alpha_exp,
                                  float* __restrict__ denom, int e_cnt)
{
    int e = blockIdx.x * blockDim.x + threadIdx.x;
    if (e >= e_cnt) return;
    const int s = src[e], d = dst[e];
    const float ea = edge_attr[e];
#pragma unroll
    for (int h = 0; h < HEADS; ++h) {
        float a = a_src[s * HEADS + h] + a_dst[d * HEADS + h];
        a = (a > 0.f) ? a : 0.2f * a;             // leaky_relu(., 0.2)
        a += ea * eW[h] + eb[h];
        a = fminf(5.f, fmaxf(-5.f, a));
        const float ex = expf(a);
        alpha_exp[(size_t)e * HEADS + h] = ex;
        atomicAdd(&denom[d * HEADS + h], ex);
    }
}

// ---------------------------------------------------------------------------
// One wave32 per edge: gather x_src[src], scale per head, atomic scatter-add.
// Lane l handles feature h*32+l of each head -> coalesced 32-wide accesses.
// ---------------------------------------------------------------------------
__global__ void scatter_msgs_kernel(const int* __restrict__ src,
                                    const int* __restrict__ dst,
                                    const float* __restrict__ xsrc,
                                    const float* __restrict__ alpha_exp,
                                    const float* __restrict__ denom,
                                    float* __restrict__ hnew, int e_cnt)
{
    const int lane = threadIdx.x & 31;
    const int e = (int)((blockIdx.x * (size_t)blockDim.x + threadIdx.x) >> 5);
    if (e >= e_cnt) return;
    const int s = src[e], d = dst[e];
    const float* __restrict__ xp = xsrc + (size_t)s * HID;
    float* __restrict__ op = hnew + (size_t)d * HID;
#pragma unroll
    for (int h = 0; h < HEADS; ++h) {
        const float w = alpha_exp[(size_t)e * HEADS + h]
                        / (denom[d * HEADS + h] + 1e-8f);
        atomicAdd(&op[h * HD + lane], xp[h * HD + lane] * w);
    }
}

// ---------------------------------------------------------------------------
// One wave32 per node: h = LayerNorm(gelu(hnew) + h) * ln_w + ln_b (in place).
// ---------------------------------------------------------------------------
__global__ void gelu_res_ln_kernel(const float* __restrict__ hnew,
                                   float* __restrict__ h,
                                   const float* __restrict__ ln_w,
                                   const float* __restrict__ ln_b, int n)
{
    const int lane = threadIdx.x & 31;
    const int node = (int)((blockIdx.x * (size_t)blockDim.x + threadIdx.x) >> 5);
    if (node >= n) return;
    float v[4];
    float sum = 0.f;
#pragma unroll
    for (int j = 0; j < 4; ++j) {
        const size_t ix = (size_t)node * HID + j * 32 + lane;
        float o = gelu_exact(hnew[ix]) + h[ix];
        v[j] = o;
        sum += o;
    }
#pragma unroll
    for (int off = 16; off > 0; off >>= 1) sum += __shfl_xor(sum, off, 32);
    const float mu = sum * (1.0f / HID);
    float var = 0.f;
#pragma unroll
    for (int j = 0; j < 4; ++j) { const float t = v[j] - mu; var += t * t; }
#pragma unroll
    for (int off = 16; off > 0; off >>= 1) var += __shfl_xor(var, off, 32);
    var *= (1.0f / HID);
    const float rs = rsqrtf(var + 1e-5f);
#pragma unroll
    for (int j = 0; j < 4; ++j) {
        const int c = j * 32 + lane;
        h[(size_t)node * HID + c] = (v[j] - mu) * rs * ln_w[c] + ln_b[c];
    }
}

// ---------------------------------------------------------------------------
// Mean pool stage 1: blockDim=64 (feature per thread), grid-strided nodes.
// ---------------------------------------------------------------------------
__global__ void mean_pool_kernel(const float* __restrict__ emb,
                                 float* __restrict__ gsum, int n)
{
    const int t = threadIdx.x;   // 0..63
    float s = 0.f;
    for (int node = blockIdx.x; node < n; node += gridDim.x)
        s += emb[(size_t)node * OUTD + t];
    atomicAdd(&gsum[t], s);
}

// ---------------------------------------------------------------------------
// Final MLP head: g = pool2(gelu(pool1(mean))) — one block, 128 threads.
// ---------------------------------------------------------------------------
__global__ void final_mlp_kernel(const float* __restrict__ gsum,
                                 const float* __restrict__ p1W,  // (128,64)
                                 const float* __restrict__ p1b,  // (128)
                                 const float* __restrict__ p2W,  // (64,128)
                                 const float* __restrict__ p2b,  // (64)
                                 float* __restrict__ out, int n)
{
    __shared__ float g0[OUTD];
    __shared__ float g1[HID];
    const int t = threadIdx.x;   // 0..127
    if (t < OUTD) g0[t] = gsum[t] / (float)n;
    __syncthreads();
    float s = p1b[t];
    for (int k = 0; k < OUTD; ++k) s += g0[k] * p1W[t * OUTD + k];
    g1[t] = gelu_exact(s);
    __syncthreads();
    if (t < OUTD) {
        float s2 = p2b[t];
        for (int k = 0; k < HID; ++k) s2 += g1[k] * p2W[t * HID + k];
        out[t] = s2;
    }
}

// ---------------------------------------------------------------------------
// Host orchestration.
// Input flattening (JAX pytree: dict keys sorted inside params):
//   0 node_feat  1 edge_index  2 edge_attr
//   per layer l (base 3 + 10*l):
//     +0 attn_dst +1 attn_src +2 dst.W +3 dst.b +4 edge.W +5 edge.b
//     +6 ln_b     +7 ln_w     +8 src.W +9 src.b
//   33 node_out.W 34 node_out.b 35 node_proj.W 36 node_proj.b
//   37 pool1.W 38 pool1.b 39 pool2.W 40 pool2.b
// ---------------------------------------------------------------------------
extern "C" void kernel_launch(void* const* d_in, const int* in_sizes, int n_in,
                              void* d_out, int out_size, void* d_ws, size_t ws_size,
                              hipStream_t stream)
{
    const float* node_feat = (const float*)d_in[0];
    const int*   src       = (const int*)d_in[1];
    const int*   dst       = src + GE;
    const float* edge_attr = (const float*)d_in[2];

    const float* node_out_W  = (const float*)d_in[33];
    const float* node_out_b  = (const float*)d_in[34];
    const float* node_proj_W = (const float*)d_in[35];
    const float* node_proj_b = (const float*)d_in[36];
    const float* pool1_W     = (const float*)d_in[37];
    const float* pool1_b     = (const float*)d_in[38];
    const float* pool2_W     = (const float*)d_in[39];
    const float* pool2_b     = (const float*)d_in[40];

    // workspace carve-up (floats)
    float* ws    = (float*)d_ws;
    float* h     = ws;                      // N*HID
    float* hnew  = h     + (size_t)GN * HID;
    float* xsrc  = hnew  + (size_t)GN * HID;
    float* xdst  = xsrc  + (size_t)GN * HID;
    float* a_src = xdst  + (size_t)GN * HID;   // N*HEADS
    float* a_dst = a_src + (size_t)GN * HEADS;
    float* denom = a_dst + (size_t)GN * HEADS;
    float* alpha = denom + (size_t)GN * HEADS; // E*HEADS
    float* gsum  = alpha + (size_t)GE * HEADS; // OUTD

    float* out_emb = (float*)d_out;            // N*OUTD
    float* out_g   = out_emb + (size_t)GN * OUTD;

    const dim3 gemmBlk(32, 4);                 // 4 waves = 64 rows per block
    const int rowBlocks = (GN + 63) / 64;      // 782 (tail handled in-kernel)
    const size_t lds_h128 = (size_t)HID  * (HID  + 4) * sizeof(float); // 67.6KB
    const size_t lds_proj = (size_t)HID  * (F_IN + 4) * sizeof(float); // 18.4KB
    const size_t lds_out  = (size_t)OUTD * (HID  + 4) * sizeof(float); // 33.8KB

    // 1) h = gelu(node_feat @ Wp^T + bp)
    wmma_gemm_lds<8><<<rowBlocks, gemmBlk, lds_proj, stream>>>(
        node_feat, node_proj_W, node_proj_b, h, GN, F_IN, 1);

    for (int l = 0; l < NLAYERS; ++l) {
        const int base = 3 + 10 * l;
        const float* attn_dst = (const float*)d_in[base + 0];
        const float* attn_src = (const float*)d_in[base + 1];
        const float* dstW     = (const float*)d_in[base + 2];
        const float* dstb     = (const float*)d_in[base + 3];
        const float* edgeW    = (const float*)d_in[base + 4];
        const float* edgeb    = (const float*)d_in[base + 5];
        const float* ln_b     = (const float*)d_in[base + 6];
        const float* ln_w     = (const float*)d_in[base + 7];
        const float* srcW     = (const float*)d_in[base + 8];
        const float* srcb     = (const float*)d_in[base + 9];

        // 2a/2b) projections
        wmma_gemm_lds<8><<<rowBlocks, gemmBlk, lds_h128, stream>>>(
            h, srcW, srcb, xsrc, GN, HID, 0);
        wmma_gemm_lds<8><<<rowBlocks, gemmBlk, lds_h128, stream>>>(
            h, dstW, dstb, xdst, GN, HID, 0);

        // 2c) per-node logits
        attn_logits_kernel<<<(GN * HEADS + 255) / 256, 256, 0, stream>>>(
            xsrc, xdst, attn_src, attn_dst, a_src, a_dst, GN);

        // 2d) zero accumulators
        zero_f32<<<(GN * HEADS + 255) / 256, 256, 0, stream>>>(denom, (size_t)GN * HEADS);
        zero_f32<<<((size_t)GN * HID + 255) / 256, 256, 0, stream>>>(hnew, (size_t)GN * HID);

        // 2e) edge softmax numerator + denominator segment-sum
        edge_alpha_kernel<<<(GE + 255) / 256, 256, 0, stream>>>(
            src, dst, edge_attr, edgeW, edgeb, a_src, a_dst, alpha, denom, GE);

        // 2f) one wave per edge: weighted message scatter
        scatter_msgs_kernel<<<(GE + 7) / 8, 256, 0, stream>>>(
            src, dst, xsrc, alpha, denom, hnew, GE);

        // 2g) gelu + residual + layernorm (in place on h)
        gelu_res_ln_kernel<<<(GN + 7) / 8, 256, 0, stream>>>(hnew, h, ln_w, ln_b, GN);
    }

    // 3) node_embeds -> d_out directly
    wmma_gemm_lds<4><<<rowBlocks, gemmBlk, lds_out, stream>>>(
        h, node_out_W, node_out_b, out_emb, GN, HID, 0);

    // 4-6) pooled head
    zero_f32<<<1, 64, 0, stream>>>(gsum, OUTD);
    mean_pool_kernel<<<128, 64, 0, stream>>>(out_emb, gsum, GN);
    final_mlp_kernel<<<1, HID, 0, stream>>>(gsum, pool1_W, pool1_b,
                                            pool2_W, pool2_b, out_g, GN);
}